// BertCRFForTokenClassification_64407329571688
// MI455X (gfx1250) — compile-verified
//
#include <hip/hip_runtime.h>
#include <hip/hip_bf16.h>
#include <math.h>

// ---------------------------------------------------------------------------
// BertCRF token classification for gfx1250 (MI455X).
//  Phase 1: logits = hidden @ W + b via V_WMMA_F32_16X16X4_F32 (fp32 WMMA),
//           bandwidth-bound (100.7 MB hidden reads -> ~4.3us at 23.3 TB/s).
//  Phase 2: per-batch CRF forward (logZ), numerator, Viterbi + backtrace.
//           One wave32 per batch element; lane j owns label j.
//  Phase 3: loss = -mean(num - logZ).
// ---------------------------------------------------------------------------

typedef __attribute__((ext_vector_type(2))) float v2f;
typedef __attribute__((ext_vector_type(8))) float v8f;

#define BB 64
#define SS 512
#define HH 768
#define LL 9
#define LPAD 16
#define KSTEPS (HH / 4)     // 192 WMMA k-steps of 4
#define NROWS (BB * SS)     // 32768 output rows

// ---------------------------------------------------------------------------
// Kernel 1: logits GEMM. 8 waves/block, each wave computes a 16-row x 16-col
// tile (cols 9..15 are zero-padded W columns). B fragments for the whole
// K-dim are pre-packed into LDS in the exact WMMA per-lane layout.
// ---------------------------------------------------------------------------
__global__ __launch_bounds__(256) void logits_wmma_kernel(
    const float* __restrict__ hs,     // (B*S, 768)
    const float* __restrict__ W,      // (768, 9)
    const float* __restrict__ bias,   // (9,)
    float* __restrict__ logits)       // (B*S, 16) padded
{
    __shared__ float bfrag[KSTEPS * 32 * 2];   // 48 KB: [k0idx][lane][2]

    const int tid = threadIdx.x;

    // Pack B fragments: lane needs W[kk][n], W[kk+1][n] with
    // n = lane%16, kk = 4*k0idx + 2*(lane/16). Zero-pad n >= 9.
    for (int idx = tid; idx < KSTEPS * 32; idx += 256) {
        const int k0idx = idx >> 5;
        const int ln    = idx & 31;
        const int n     = ln & 15;
        const int kk    = k0idx * 4 + 2 * (ln >> 4);
        float bx = 0.0f, by = 0.0f;
        if (n < LL) {
            bx = W[kk * LL + n];
            by = W[(kk + 1) * LL + n];
        }
        bfrag[idx * 2 + 0] = bx;
        bfrag[idx * 2 + 1] = by;
    }
    __syncthreads();

    const int lane    = tid & 31;
    const int warp    = tid >> 5;
    const int rowBase = (blockIdx.x * 8 + warp) * 16;
    const int m       = lane & 15;          // A row within tile
    const int khalf   = (lane >> 4) * 2;    // K sub-offset 0 or 2

    const float* aptr = hs + (size_t)(rowBase + m) * HH + khalf;

    v8f acc = {};
    #pragma unroll 4
    for (int k0 = 0; k0 < KSTEPS; ++k0) {
        v2f a  = *(const v2f*)(aptr + k0 * 4);                   // global_load_b64
        v2f bv = *(const v2f*)(&bfrag[(k0 * 32 + lane) * 2]);    // ds_load_b64
        acc = __builtin_amdgcn_wmma_f32_16x16x4_f32(
            /*neg_a=*/false, a, /*neg_b=*/false, bv,
            /*c_mod=*/(short)0, acc, /*reuse_a=*/false, /*reuse_b=*/false);
    }

    // D layout: c[i] = D[row = i + 8*(lane/16)][col = lane%16]
    const int   n    = lane & 15;
    const float badd = (n < LL) ? bias[n] : 0.0f;
    const int   mOff = (lane >> 4) * 8;
    #pragma unroll
    for (int i = 0; i < 8; ++i) {
        logits[(size_t)(rowBase + mOff + i) * LPAD + n] = acc[i] + badd;
    }
}

// ---------------------------------------------------------------------------
// Kernel 2: CRF per batch element. 1 wave32 per block, grid = 64.
// Lane j (j<9) owns label j: transition column T[:,j] in registers.
// Forward logsumexp scan + Viterbi scan fused in one t-loop.
// ---------------------------------------------------------------------------
__global__ __launch_bounds__(32) void crf_kernel(
    const float* __restrict__ logits,   // (B*S, 16)
    const float* __restrict__ trans,    // (9,9)
    const float* __restrict__ startT,   // (9,)
    const float* __restrict__ endT,     // (9,)
    const int*   __restrict__ attn,     // (B,S)
    const int*   __restrict__ labels,   // (B,S)
    float* __restrict__ diff,           // (B,) num - logZ
    float* __restrict__ preds)          // (B,S) as float
{
    const int  b    = blockIdx.x;
    const int  lane = threadIdx.x;
    const bool act  = lane < LL;
    const float NEG = -1e30f;

    __shared__ unsigned char bps[SS * LL];   // backpointers, t = 1..511

    float Tc[LL];
    #pragma unroll
    for (int i = 0; i < LL; ++i) Tc[i] = trans[i * LL + (act ? lane : 0)];

    const float* lg = logits + (size_t)b * SS * LPAD;
    const int*   am = attn + b * SS;
    const int*   lb = labels + b * SS;

    float emit0  = act ? lg[lane] : NEG;
    float alpha  = act ? (startT[act ? lane : 0] + emit0) : NEG;
    float vscore = alpha;

    for (int t = 1; t < SS; ++t) {
        const float e   = act ? lg[t * LPAD + lane] : NEG;
        const int   lab = lb[t];
        const bool  mk  = am[t] != 0;               // attention mask
        const bool  cm  = (lab != -100) && mk;      // crf mask

        float av[LL], vv[LL];
        #pragma unroll
        for (int i = 0; i < LL; ++i) {
            av[i] = __shfl(alpha, i, 32);
            vv[i] = __shfl(vscore, i, 32);
        }

        // forward: nxt[j] = emit[j] + logsumexp_i(alpha[i] + T[i][j])
        float mx = NEG;
        #pragma unroll
        for (int i = 0; i < LL; ++i) mx = fmaxf(mx, av[i] + Tc[i]);
        float s = 0.0f;
        #pragma unroll
        for (int i = 0; i < LL; ++i) s += __expf(av[i] + Tc[i] - mx);
        const float nxt = e + mx + __logf(s);
        if (act && cm) alpha = nxt;

        // viterbi: best over i of vscore[i] + T[i][j] (first-max ties)
        float bv = NEG; int bi = 0;
        #pragma unroll
        for (int i = 0; i < LL; ++i) {
            const float c = vv[i] + Tc[i];
            if (c > bv) { bv = c; bi = i; }
        }
        if (act) bps[t * LL + lane] = (unsigned char)(mk ? bi : lane);
        if (act && mk) vscore = bv + e;
    }
    __syncthreads();

    // logZ = logsumexp_j(alpha[j] + end[j])
    const float zv = act ? alpha + endT[lane] : NEG;
    float zmax = zv;
    #pragma unroll
    for (int off = 16; off; off >>= 1) zmax = fmaxf(zmax, __shfl_xor(zmax, off, 32));
    float zs = act ? __expf(zv - zmax) : 0.0f;
    #pragma unroll
    for (int off = 16; off; off >>= 1) zs += __shfl_xor(zs, off, 32);
    const float logZ = zmax + __logf(zs);

    // viterbi terminal: argmax_j(vscore[j] + end[j]) (first-max ties)
    float fvv = act ? vscore + endT[lane] : NEG;
    int   fii = act ? lane : 1000;
    #pragma unroll
    for (int off = 16; off; off >>= 1) {
        const float ov = __shfl_xor(fvv, off, 32);
        const int   oi = __shfl_xor(fii, off, 32);
        if (ov > fvv || (ov == fvv && oi < fii)) { fvv = ov; fii = oi; }
    }
    const int last = fii;

    // numerator: parallel over t, strided per lane
    float np = 0.0f; int cp = 0;
    for (int t = lane; t < SS; t += 32) {
        const int  lab = lb[t];
        const int  tag = (lab == -100) ? 0 : lab;
        const bool mk  = am[t] != 0;
        const bool cm  = ((lab != -100) && mk) || (t == 0);
        cp += cm ? 1 : 0;
        const float e = lg[t * LPAD + tag];
        if (t == 0) {
            np += startT[tag] + e;
        } else if (cm) {
            const int labp = lb[t - 1];
            const int tagp = (labp == -100) ? 0 : labp;
            np += trans[tagp * LL + tag] + e;
        }
    }
    #pragma unroll
    for (int off = 16; off; off >>= 1) {
        np += __shfl_xor(np, off, 32);
        cp += __shfl_xor(cp, off, 32);
    }
    const int last_idx = cp - 1;
    const int lastlab  = lb[last_idx];
    const int last_tag = (lastlab == -100) ? 0 : lastlab;
    const float num    = np + endT[last_tag];

    if (lane == 0) diff[b] = num - logZ;

    // backtrace (lane 0): tags_all[511]=last; tags_all[t-1]=bps[t][tags_all[t]]
    if (lane == 0) {
        int tag = last;
        for (int t = SS - 1; t >= 1; --t) {
            preds[b * SS + t] = am[t] ? (float)tag : -100.0f;
            tag = bps[t * LL + tag];
        }
        preds[b * SS + 0] = am[0] ? (float)tag : -100.0f;
    }
}

// ---------------------------------------------------------------------------
// Kernel 3: loss = -mean(diff)
// ---------------------------------------------------------------------------
__global__ __launch_bounds__(64) void loss_kernel(const float* __restrict__ diff,
                                                  float* __restrict__ out)
{
    const int tid = threadIdx.x;
    __shared__ float smem[64];
    smem[tid] = diff[tid];
    __syncthreads();
    if (tid < 32) {
        float x = smem[tid] + smem[tid + 32];
        #pragma unroll
        for (int off = 16; off; off >>= 1) x += __shfl_xor(x, off, 32);
        if (tid == 0) out[0] = -x / (float)BB;
    }
}

extern "C" void kernel_launch(void* const* d_in, const int* in_sizes, int n_in,
                              void* d_out, int out_size, void* d_ws, size_t ws_size,
                              hipStream_t stream) {
    const float* hs     = (const float*)d_in[0];   // hidden_states
    const float* W      = (const float*)d_in[1];   // W
    const float* bias   = (const float*)d_in[2];   // b
    const float* startT = (const float*)d_in[3];   // start_transitions
    const float* endT   = (const float*)d_in[4];   // end_transitions
    const float* trans  = (const float*)d_in[5];   // transitions
    const int*   attn   = (const int*)d_in[6];     // attention_mask
    const int*   labels = (const int*)d_in[7];     // labels

    float* out    = (float*)d_out;                 // [0]=loss, [1..]=preds
    float* logits = (float*)d_ws;                  // (32768, 16) = 2 MB
    float* diff   = logits + (size_t)NROWS * LPAD; // (64,)

    // 2048 row-tiles of 16, 8 waves per block -> 256 blocks
    logits_wmma_kernel<<<256, 256, 0, stream>>>(hs, W, bias, logits);
    crf_kernel<<<BB, 32, 0, stream>>>(logits, trans, startT, endT, attn, labels,
                                      diff, out + 1);
    loss_kernel<<<1, 64, 0, stream>>>(diff, out);
}